// QuanvolutionAdvanced_65481071399030
// MI455X (gfx1250) — compile-verified
//
#include <hip/hip_runtime.h>
#include <math.h>

// ---------------------------------------------------------------------------
// Fused "quanvolution + tiny MHA + classifier" for MI455X (gfx1250, wave32).
// One workgroup (256 threads = 8 wave32) per batch element. All intermediates
// live in LDS (~189 KB of the 320 KB WGP budget). Attention matmuls use
// V_WMMA_F32_16X16X4_F32 (fp32 matrix pipe, matches reference precision).
// Softmax normalization is deferred to the O-store (flash-attention style).
// ---------------------------------------------------------------------------

typedef float v2f __attribute__((ext_vector_type(2)));
typedef float v8f __attribute__((ext_vector_type(8)));

#define LSEQ   196   // real sequence length (14*14 patches)
#define LPAD   208   // padded to 13 tiles of 16
#define NT     13    // tiles along L
#define SROW   212   // score-matrix row stride (floats); keeps rows 16B-aligned
#define RSQRT2 0.70710678118654752f

__device__ __forceinline__ void two_qubit(float xa, float xb, float pa, float pb,
                                          float cg, float sg, float ch, float sh,
                                          float& z0, float& z1) {
    // Analytic 2-qubit Ry/CNOT block (real amplitudes), matches reference.
    const float ha = 0.5f * (xa + pa), hb = 0.5f * (xb + pb);
    const float ca = cosf(ha), sa = sinf(ha);
    const float cb = cosf(hb), sb = sinf(hb);
    float p00 = ca * cb, p01 = ca * sb, p10 = sa * cb, p11 = sa * sb;
    // CNOT(0,1): swap |10> <-> |11>
    float t = p10; p10 = p11; p11 = t;
    const float q00 = cg * p00 - sg * p10;
    const float q10 = sg * p00 + cg * p10;
    const float q01 = cg * p01 - sg * p11;
    const float q11 = sg * p01 + cg * p11;
    const float r00 = ch * q00 - sh * q01;
    const float r01 = sh * q00 + ch * q01;
    const float r10 = ch * q10 - sh * q11;
    const float r11 = sh * q10 + ch * q11;
    const float a00 = r00 * r00, a01 = r01 * r01, a10 = r10 * r10, a11 = r11 * r11;
    z0 = a00 + a01 - a10 - a11;
    z1 = a00 - a01 + a10 - a11;
}

__global__ __launch_bounds__(256) void quanv_attn_kernel(
    const float* __restrict__ x,        // (B,784)
    const float* __restrict__ qparams,  // (8)
    const float* __restrict__ in_w,     // (12,4)
    const float* __restrict__ in_b,     // (12)
    const float* __restrict__ out_w,    // (4,4)
    const float* __restrict__ out_b,    // (4)
    const float* __restrict__ ln_w,     // (4)
    const float* __restrict__ ln_b,     // (4)
    const float* __restrict__ lin_w,    // (10,784)
    const float* __restrict__ lin_b,    // (10)
    float* __restrict__ out)            // (B,10)
{
    extern __shared__ float smem[];
    float* s_p   = smem;                  // SROW * LPAD   scores / exp (reused per head)
    float* s_q   = s_p + SROW * LPAD;     // LPAD*4  (scaled by 1/sqrt(2))
    float* s_k   = s_q + LPAD * 4;        // LPAD*4
    float* s_v   = s_k + LPAD * 4;        // LPAD*4
    float* s_oat = s_v + LPAD * 4;        // LSEQ*4  attention output (both heads)
    float* s_img = s_oat + LSEQ * 4;      // 784     image, later reused for LN output
    float* s_inv = s_img + 784;           // LPAD    per-row 1/sum(exp)

    __shared__ float sw_in[48], sb_in[12], sw_out[16], sb_out[4];
    __shared__ float s_lnw[4], s_lnb[4], s_qp[8], s_linb[10];
    __shared__ float s_tgc[4], s_tgs[4];  // cos/sin(0.5*qparams[4..7])

    const int tid  = threadIdx.x;
    const int lane = tid & 31;
    const int wave = tid >> 5;
    const int b    = blockIdx.x;

    // ---- stage small params + image; warm classifier weights in cache ----
    if (tid < 48) sw_in[tid] = in_w[tid];
    if (tid < 12) sb_in[tid] = in_b[tid];
    if (tid < 16) sw_out[tid] = out_w[tid];
    if (tid < 4)  {
        sb_out[tid] = out_b[tid]; s_lnw[tid] = ln_w[tid]; s_lnb[tid] = ln_b[tid];
        const float g = 0.5f * qparams[4 + tid];
        s_tgc[tid] = cosf(g); s_tgs[tid] = sinf(g);
    }
    if (tid < 8)  s_qp[tid] = qparams[tid];
    if (tid < 10) s_linb[tid] = lin_b[tid];
    if (tid < 245) __builtin_prefetch(lin_w + tid * 32, 0, 3);  // 7840 floats of lin_w
    for (int idx = tid; idx < 784; idx += 256) s_img[idx] = x[b * 784 + idx];
    __syncthreads();

    // ---- quantum feature map + qkv projection (one thread per position) ----
    if (tid < LSEQ) {
        const int pr = tid / 14, pc = tid % 14;
        const float xa = s_img[(2 * pr) * 28 + 2 * pc];
        const float xb = s_img[(2 * pr) * 28 + 2 * pc + 1];
        const float xc = s_img[(2 * pr + 1) * 28 + 2 * pc];
        const float xd = s_img[(2 * pr + 1) * 28 + 2 * pc + 1];
        float z0, z1, z2, z3;
        two_qubit(xa, xb, s_qp[0], s_qp[1], s_tgc[0], s_tgs[0], s_tgc[1], s_tgs[1], z0, z1);
        two_qubit(xc, xd, s_qp[2], s_qp[3], s_tgc[2], s_tgs[2], s_tgc[3], s_tgs[3], z2, z3);
        #pragma unroll
        for (int c = 0; c < 4; ++c) {
            const float qv = sb_in[c]     + z0 * sw_in[c * 4]           + z1 * sw_in[c * 4 + 1]
                                          + z2 * sw_in[c * 4 + 2]       + z3 * sw_in[c * 4 + 3];
            const float kv = sb_in[c + 4] + z0 * sw_in[(c + 4) * 4]     + z1 * sw_in[(c + 4) * 4 + 1]
                                          + z2 * sw_in[(c + 4) * 4 + 2] + z3 * sw_in[(c + 4) * 4 + 3];
            const float vv = sb_in[c + 8] + z0 * sw_in[(c + 8) * 4]     + z1 * sw_in[(c + 8) * 4 + 1]
                                          + z2 * sw_in[(c + 8) * 4 + 2] + z3 * sw_in[(c + 8) * 4 + 3];
            s_q[tid * 4 + c] = qv * RSQRT2;   // fold 1/sqrt(head_dim) into Q
            s_k[tid * 4 + c] = kv;
            s_v[tid * 4 + c] = vv;
        }
    } else if (tid < LPAD) {                  // zero the L padding rows
        #pragma unroll
        for (int c = 0; c < 4; ++c) {
            s_q[tid * 4 + c] = 0.0f; s_k[tid * 4 + c] = 0.0f; s_v[tid * 4 + c] = 0.0f;
        }
    }
    __syncthreads();

    const int  l16 = lane & 15;
    const bool lo  = lane < 16;

    // ---- attention, one head at a time (score buffer reused) ----
    for (int h = 0; h < 2; ++h) {
        // S = Q_h * K_h^T : 13x13 tiles, one V_WMMA_F32_16X16X4_F32 each (K: 2 padded to 4).
        // A layout: lanes 0-15 hold K=0,1 in v[0],v[1]; lanes 16-31 hold K=2,3 (zeros here).
        // B layout (4x16): v[0] = rows K=0 (lanes 0-15) / K=2 (lanes 16-31); v[1] = K=1 / K=3.
        // Branch-free operand fetch: unconditional ds_load_b64 + v_cndmask to zero.
        for (int t = wave; t < NT * NT; t += 8) {
            const int i = t / NT, j = t % NT;
            const v2f qa = *(const v2f*)&s_q[(16 * i + l16) * 4 + 2 * h];
            const v2f ka = *(const v2f*)&s_k[(16 * j + l16) * 4 + 2 * h];
            v2f a, bb;
            a.x  = lo ? qa.x : 0.0f;  a.y  = lo ? qa.y : 0.0f;
            bb.x = lo ? ka.x : 0.0f;  bb.y = lo ? ka.y : 0.0f;
            v8f acc = {};
            acc = __builtin_amdgcn_wmma_f32_16x16x4_f32(
                      false, a, false, bb, (short)0, acc, false, false);
            // D layout: lane 0-15 -> M=r, lane 16-31 -> M=r+8; N = lane&15
            const int row0 = 16 * i + (lo ? 0 : 8);
            const int col  = 16 * j + l16;
            #pragma unroll
            for (int r = 0; r < 8; ++r) s_p[(row0 + r) * SROW + col] = acc[r];
        }
        __syncthreads();

        // Row softmax (one thread per row), float4-vectorized; normalization is
        // deferred: s_p keeps unnormalized exp, s_inv[row] = 1/sum.
        if (tid < LSEQ) {
            float4* row4 = (float4*)(s_p + tid * SROW);   // 16B-aligned (SROW%4==0)
            float m = -1e30f;
            for (int c4 = 0; c4 < LSEQ / 4; ++c4) {
                const float4 v = row4[c4];
                m = fmaxf(m, fmaxf(fmaxf(v.x, v.y), fmaxf(v.z, v.w)));
            }
            float s = 0.0f;
            for (int c4 = 0; c4 < LSEQ / 4; ++c4) {
                float4 v = row4[c4];
                v.x = __expf(v.x - m); v.y = __expf(v.y - m);
                v.z = __expf(v.z - m); v.w = __expf(v.w - m);
                row4[c4] = v;
                s += (v.x + v.y) + (v.z + v.w);
            }
            s_inv[tid] = 1.0f / s;
            // cols 196..207 keep stale scores: they multiply zero V rows below
        }
        __syncthreads();

        // O = P * V_h : per 16-row tile, 52 chained K=4 WMMAs (K spans 208).
        const int n    = l16;
        const int nc   = (n < 2) ? n : 0;          // clamped (always-valid) V column
        const int koff = lo ? 0 : 2;
        for (int i = wave; i < NT; i += 8) {
            v8f acc = {};
            const int arow = 16 * i + l16;
            #pragma unroll 4
            for (int kk = 0; kk < LPAD / 4; ++kk) {
                const v2f a = *(const v2f*)&s_p[arow * SROW + 4 * kk + koff];  // ds_load_b64
                const float bx = s_v[(4 * kk + koff) * 4 + 2 * h + nc];
                const float by = s_v[(4 * kk + koff + 1) * 4 + 2 * h + nc];
                v2f bb;
                bb.x = (n < 2) ? bx : 0.0f;
                bb.y = (n < 2) ? by : 0.0f;
                acc = __builtin_amdgcn_wmma_f32_16x16x4_f32(
                          false, a, false, bb, (short)0, acc, false, false);
            }
            const int row0 = 16 * i + (lo ? 0 : 8);
            if (n < 2) {
                #pragma unroll
                for (int r = 0; r < 8; ++r) {
                    const int row = row0 + r;
                    if (row < LSEQ)
                        s_oat[row * 4 + 2 * h + n] = acc[r] * s_inv[row];  // deferred softmax scale
                }
            }
        }
        __syncthreads();
    }

    // ---- out_proj + LayerNorm (one thread per position); write into s_img ----
    if (tid < LSEQ) {
        const float o0 = s_oat[tid * 4], o1 = s_oat[tid * 4 + 1];
        const float o2 = s_oat[tid * 4 + 2], o3 = s_oat[tid * 4 + 3];
        float y[4];
        #pragma unroll
        for (int e = 0; e < 4; ++e)
            y[e] = sb_out[e] + o0 * sw_out[e * 4]     + o1 * sw_out[e * 4 + 1]
                             + o2 * sw_out[e * 4 + 2] + o3 * sw_out[e * 4 + 3];
        const float mu = 0.25f * (y[0] + y[1] + y[2] + y[3]);
        const float d0 = y[0] - mu, d1 = y[1] - mu, d2 = y[2] - mu, d3 = y[3] - mu;
        const float var = 0.25f * (d0 * d0 + d1 * d1 + d2 * d2 + d3 * d3);
        const float inv = rsqrtf(var + 1e-5f);
        #pragma unroll
        for (int e = 0; e < 4; ++e)
            s_img[tid * 4 + e] = (y[e] - mu) * inv * s_lnw[e] + s_lnb[e];
    }
    __syncthreads();

    // ---- classifier: logits[j] = sum_{c,l} ln[l][c] * lin_w[j][c*196+l] ----
    float part[10];
    #pragma unroll
    for (int j = 0; j < 10; ++j) part[j] = 0.0f;
    for (int idx = tid; idx < 784; idx += 256) {     // idx = c*196 + l (coalesced over threads)
        const int c = idx / 196, l = idx % 196;
        const float val = s_img[l * 4 + c];
        #pragma unroll
        for (int j = 0; j < 10; ++j) part[j] = fmaf(val, lin_w[j * 784 + idx], part[j]);
    }
    // deterministic tree reduction in the (now free) score buffer
    float* red = s_p;                                 // needs 10*256 floats
    #pragma unroll
    for (int j = 0; j < 10; ++j) red[j * 256 + tid] = part[j];
    __syncthreads();
    for (int s2 = 128; s2 > 0; s2 >>= 1) {
        if (tid < s2) {
            #pragma unroll
            for (int j = 0; j < 10; ++j) red[j * 256 + tid] += red[j * 256 + tid + s2];
        }
        __syncthreads();
    }
    if (tid == 0) {
        float lg[10];
        float m = -1e30f;
        #pragma unroll
        for (int j = 0; j < 10; ++j) { lg[j] = red[j * 256] + s_linb[j]; m = fmaxf(m, lg[j]); }
        float s = 0.0f;
        #pragma unroll
        for (int j = 0; j < 10; ++j) s += __expf(lg[j] - m);
        const float lse = m + __logf(s);
        #pragma unroll
        for (int j = 0; j < 10; ++j) out[b * 10 + j] = lg[j] - lse;
    }
}

extern "C" void kernel_launch(void* const* d_in, const int* in_sizes, int n_in,
                              void* d_out, int out_size, void* d_ws, size_t ws_size,
                              hipStream_t stream) {
    (void)n_in; (void)d_ws; (void)ws_size; (void)out_size;
    const float* x   = (const float*)d_in[0];
    const float* qp  = (const float*)d_in[1];
    const float* inw = (const float*)d_in[2];
    const float* inb = (const float*)d_in[3];
    const float* ow  = (const float*)d_in[4];
    const float* ob  = (const float*)d_in[5];
    const float* lnw = (const float*)d_in[6];
    const float* lnb = (const float*)d_in[7];
    const float* lw  = (const float*)d_in[8];
    const float* lb  = (const float*)d_in[9];
    float* out = (float*)d_out;

    const int B = in_sizes[0] / 784;
    const size_t shmem =
        (size_t)(SROW * LPAD + 3 * LPAD * 4 + LSEQ * 4 + 784 + LPAD) * sizeof(float);
    quanv_attn_kernel<<<B, 256, shmem, stream>>>(x, qp, inw, inb, ow, ob,
                                                 lnw, lnb, lw, lb, out);
}